// EuclideanDistanceLoss_55190329754162
// MI455X (gfx1250) — compile-verified
//
#include <hip/hip_runtime.h>
#include <math.h>

typedef float v2f __attribute__((ext_vector_type(2)));
typedef float v8f __attribute__((ext_vector_type(8)));

#define KDIM 16384            // 128*128 flattened
#define NMAT 1024             // number of matrices

// Wave tile: 32 (M) x 64 (N). Block: 2x2 waves -> 64 x 128 block tile.
#define WTM 32
#define WTN 64
#define BTM 64
#define BTN 128
#define BLOCKS_I (NMAT / BTM)                  // 16
#define BLOCKS_J (NMAT / BTN)                  // 8
#define NBLOCKS (BLOCKS_I * BLOCKS_J)          // 128
#define WAVES_PER_BLOCK 4
#define NPARTIALS (NBLOCKS * WAVES_PER_BLOCK)  // 512

// ws layout (floats): [0,1024) csq | [1024,2048) dsq | [2048,2048+NPARTIALS) partials

#define WMMA_F32(a, b, c) \
    __builtin_amdgcn_wmma_f32_16x16x4_f32(false, (a), false, (b), (short)0, (c), false, false)

// ---------------- row squared norms ----------------
__global__ void __launch_bounds__(256)
sqnorm_kernel(const float* __restrict__ X, float* __restrict__ out) {
    const int row = blockIdx.x;
    const float4* p = (const float4*)(X + (size_t)row * KDIM);
    float s = 0.f;
    for (int i = threadIdx.x; i < KDIM / 4; i += 256) {
        float4 v = p[i];
        s += v.x * v.x + v.y * v.y + v.z * v.z + v.w * v.w;
    }
    __shared__ float red[256];
    red[threadIdx.x] = s;
    __syncthreads();
    for (int off = 128; off > 0; off >>= 1) {
        if (threadIdx.x < off) red[threadIdx.x] += red[threadIdx.x + off];
        __syncthreads();
    }
    if (threadIdx.x == 0) out[row] = red[0];
}

// ---------------- fused gram tile + distance + partial sum ----------------
__device__ __forceinline__ float epilogue8(v8f acc, const float* __restrict__ csq,
                                           int mbase, float dn) {
    float s = 0.f;
#pragma unroll
    for (int r = 0; r < 8; ++r) {
        float d2 = csq[mbase + r] + dn - 2.0f * acc[r];
        s += sqrtf(fmaxf(d2, 0.0f));
    }
    return s;
}

__global__ void __launch_bounds__(32 * WAVES_PER_BLOCK)
gram_tile_kernel(const float* __restrict__ Cf, const float* __restrict__ Df,
                 const float* __restrict__ csq, const float* __restrict__ dsq,
                 float* __restrict__ partials) {
    const int lane = threadIdx.x & 31;
    const int wave = threadIdx.x >> 5;

    // 2x2 wave arrangement inside the block tile
    const int wi = wave >> 1;             // 0..1
    const int wj = wave & 1;              // 0..1
    const int bi = (blockIdx.x / BLOCKS_J) * BTM;
    const int bj = (blockIdx.x % BLOCKS_J) * BTN;
    const int i0 = bi + wi * WTM;
    const int j0 = bj + wj * WTN;

    // A (16x4 f32, M x K) and B (4x16 f32, K x N with B = Df^T tile) share the
    // same per-lane pattern: 2 consecutive floats from row (lane&15),
    // col k + 2*(lane>>4)  -> one 8B-aligned global_load_b64 per fragment.
    const int frow = lane & 15;
    const int koff = (lane >> 4) << 1;

    const float* a0p = Cf + (size_t)(i0 + frow) * KDIM + koff;
    const float* a1p = a0p + (size_t)16 * KDIM;
    const float* b0p = Df + (size_t)(j0 + frow) * KDIM + koff;
    const float* b1p = b0p + (size_t)16 * KDIM;
    const float* b2p = b0p + (size_t)32 * KDIM;
    const float* b3p = b0p + (size_t)48 * KDIM;

    v8f acc00 = {}, acc01 = {}, acc02 = {}, acc03 = {};
    v8f acc10 = {}, acc11 = {}, acc12 = {}, acc13 = {};

#pragma unroll 4
    for (int k = 0; k < KDIM; k += 4) {
        v2f a0 = *(const v2f*)(a0p + k);
        v2f a1 = *(const v2f*)(a1p + k);
        v2f b0 = *(const v2f*)(b0p + k);
        v2f b1 = *(const v2f*)(b1p + k);
        v2f b2 = *(const v2f*)(b2p + k);
        v2f b3 = *(const v2f*)(b3p + k);
        acc00 = WMMA_F32(a0, b0, acc00);
        acc01 = WMMA_F32(a0, b1, acc01);
        acc02 = WMMA_F32(a0, b2, acc02);
        acc03 = WMMA_F32(a0, b3, acc03);
        acc10 = WMMA_F32(a1, b0, acc10);
        acc11 = WMMA_F32(a1, b1, acc11);
        acc12 = WMMA_F32(a1, b2, acc12);
        acc13 = WMMA_F32(a1, b3, acc13);
    }

    // Accumulator element r of a 16x16 block: (M = r + 8*(lane>>4), N = lane&15).
    const int mhalf = (lane >> 4) << 3;
    const int ncol  = lane & 15;

    const float dn0 = dsq[j0 + ncol];
    const float dn1 = dsq[j0 + 16 + ncol];
    const float dn2 = dsq[j0 + 32 + ncol];
    const float dn3 = dsq[j0 + 48 + ncol];

    float local = 0.f;
    local += epilogue8(acc00, csq, i0 + mhalf,      dn0);
    local += epilogue8(acc01, csq, i0 + mhalf,      dn1);
    local += epilogue8(acc02, csq, i0 + mhalf,      dn2);
    local += epilogue8(acc03, csq, i0 + mhalf,      dn3);
    local += epilogue8(acc10, csq, i0 + 16 + mhalf, dn0);
    local += epilogue8(acc11, csq, i0 + 16 + mhalf, dn1);
    local += epilogue8(acc12, csq, i0 + 16 + mhalf, dn2);
    local += epilogue8(acc13, csq, i0 + 16 + mhalf, dn3);

    // wave32 butterfly reduction
    for (int off = 16; off > 0; off >>= 1)
        local += __shfl_xor(local, off, 32);
    if (lane == 0) partials[blockIdx.x * WAVES_PER_BLOCK + wave] = local;
}

// ---------------- deterministic final reduction ----------------
__global__ void __launch_bounds__(256)
final_reduce_kernel(const float* __restrict__ partials, float* __restrict__ out) {
    __shared__ float red[256];
    float s = 0.f;
    for (int i = threadIdx.x; i < NPARTIALS; i += 256) s += partials[i];
    red[threadIdx.x] = s;
    __syncthreads();
    for (int off = 128; off > 0; off >>= 1) {
        if (threadIdx.x < off) red[threadIdx.x] += red[threadIdx.x + off];
        __syncthreads();
    }
    if (threadIdx.x == 0)
        out[0] = red[0] * (1.0f / ((float)NMAT * (float)NMAT));
}

extern "C" void kernel_launch(void* const* d_in, const int* in_sizes, int n_in,
                              void* d_out, int out_size, void* d_ws, size_t ws_size,
                              hipStream_t stream) {
    const float* C = (const float*)d_in[0];
    const float* D = (const float*)d_in[1];
    float* ws = (float*)d_ws;
    float* csq = ws;
    float* dsq = ws + NMAT;
    float* partials = ws + 2 * NMAT;

    sqnorm_kernel<<<NMAT, 256, 0, stream>>>(C, csq);
    sqnorm_kernel<<<NMAT, 256, 0, stream>>>(D, dsq);
    gram_tile_kernel<<<NBLOCKS, 32 * WAVES_PER_BLOCK, 0, stream>>>(
        C, D, csq, dsq, partials);
    final_reduce_kernel<<<1, 256, 0, stream>>>(partials, (float*)d_out);
}